// GFormer_29180007809574
// MI455X (gfx1250) — compile-verified
//
#include <hip/hip_runtime.h>
#include <hip/hip_bf16.h>

typedef float v2f __attribute__((ext_vector_type(2)));
typedef float v8f __attribute__((ext_vector_type(8)));

#define DIM_E 128
#define HEAD 4

// ---------------------------------------------------------------------------
// Zero-fill helper (d_out / attNorm are poisoned by the harness)
// ---------------------------------------------------------------------------
__global__ void gf_zero_f32(float* __restrict__ p, size_t n) {
    size_t i = (size_t)blockIdx.x * blockDim.x + threadIdx.x;
    size_t stride = (size_t)gridDim.x * blockDim.x;
    for (; i < n; i += stride) p[i] = 0.0f;
}

// ---------------------------------------------------------------------------
// QKV projection: O = X @ W for W in {qW,kW,vW}, fp32 via V_WMMA_F32_16X16X4_F32.
// One wave computes a full 16x128 output strip: 8 independent v8f accumulators,
// A-tile (16x128 of X) loaded exactly once per wave, 8 back-to-back WMMAs per
// K-step (independent accumulators -> no RAW hazard stalls).
// ISA 7.12.2 layouts (wave32):
//   A 16x4 fp32 : lane l holds A[l&15][ (l>>4)*2 + r ], r=0..1  (b64 load)
//   B 4x16 fp32 : lane l holds B[ (l>>4)*2 + r ][ l&15 ]
//   C/D 16x16   : lane l, reg r holds D[ (l>>4)*8 + r ][ l&15 ]
// ---------------------------------------------------------------------------
__global__ __launch_bounds__(32) void gf_qkv_gemm(
    const float* __restrict__ X,
    const float* __restrict__ qW, const float* __restrict__ kW,
    const float* __restrict__ vW,
    float* __restrict__ Q, float* __restrict__ Kout, float* __restrict__ Vout,
    int n_nodes)
{
    const int lane    = threadIdx.x;          // 0..31, EXEC all-ones for WMMA
    const int rowBase = blockIdx.x * 16;
    const int mat     = blockIdx.y;           // 0=q 1=k 2=v

    const float* W = (mat == 0) ? qW : (mat == 1) ? kW : vW;
    float*       O = (mat == 0) ? Q  : (mat == 1) ? Kout : Vout;

    const int m  = lane & 15;
    const int kk = (lane >> 4) << 1;          // 0 or 2

    // clamp row for partial tiles (loads only; stores handled below)
    int rowLd = rowBase + m;
    if (rowLd >= n_nodes) rowLd = n_nodes - 1;
    const float* xrow = X + (size_t)rowLd * DIM_E;

    v8f acc[8];
    #pragma unroll
    for (int t = 0; t < 8; ++t)
        acc[t] = (v8f){0.f, 0.f, 0.f, 0.f, 0.f, 0.f, 0.f, 0.f};

    for (int k0 = 0; k0 < DIM_E; k0 += 4) {
        // A: two contiguous fp32 -> single b64 load, shared by all 8 col tiles
        v2f a = *(const v2f*)(xrow + k0 + kk);
        const float* wr = W + (size_t)(k0 + kk) * DIM_E + m;
        #pragma unroll
        for (int t = 0; t < 8; ++t) {
            v2f b;
            b.x = wr[t * 16];                 // row k0+kk,   col 16t+m
            b.y = wr[DIM_E + t * 16];         // row k0+kk+1, col 16t+m
            acc[t] = __builtin_amdgcn_wmma_f32_16x16x4_f32(
                        /*neg_a=*/false, a, /*neg_b=*/false, b,
                        /*c_mod=*/(short)0, acc[t],
                        /*reuse_a=*/false, /*reuse_b=*/false);
        }
    }

    const int  mOut = (lane >> 4) << 3;       // 0 or 8
    const bool full = (rowBase + 16) <= n_nodes;
    #pragma unroll
    for (int t = 0; t < 8; ++t) {
        float* ocol = O + (size_t)rowBase * DIM_E + t * 16 + m;
        if (full) {
            #pragma unroll
            for (int r = 0; r < 8; ++r)
                ocol[(size_t)(mOut + r) * DIM_E] = acc[t][r];
        } else {
            #pragma unroll
            for (int r = 0; r < 8; ++r)
                if (rowBase + mOut + r < n_nodes)
                    ocol[(size_t)(mOut + r) * DIM_E] = acc[t][r];
        }
    }
}

// ---------------------------------------------------------------------------
// Edge logits: one wave per edge. Lane l owns dims [4l, 4l+4) (float4 loads,
// fully coalesced). Head h = l/8 -> 8-lane shuffle tree per head.
// expAtt goes straight into d_out's att section; softmax denominator via
// global_atomic_add_f32 into attNorm[row][head].
// ---------------------------------------------------------------------------
__global__ __launch_bounds__(256) void gf_edge_logits(
    const float* __restrict__ Q, const float* __restrict__ K,
    const int* __restrict__ rows, const int* __restrict__ cols,
    float* __restrict__ expAtt,      // [E,4] (d_out att region, raw exp here)
    float* __restrict__ attNorm,     // [N,4]
    int n_edges)
{
    const int e    = blockIdx.x * (blockDim.x >> 5) + (threadIdx.x >> 5);
    const int lane = threadIdx.x & 31;
    if (e >= n_edges) return;

    const int r = rows[e];
    const int c = cols[e];

    const float4 qa = ((const float4*)(Q + (size_t)r * DIM_E))[lane];
    const float4 ka = ((const float4*)(K + (size_t)c * DIM_E))[lane];

    float s = qa.x * ka.x + qa.y * ka.y + qa.z * ka.z + qa.w * ka.w;
    // reduce across the 8 lanes of this head
    s += __shfl_xor(s, 1, 32);
    s += __shfl_xor(s, 2, 32);
    s += __shfl_xor(s, 4, 32);

    if ((lane & 7) == 0) {
        const int h = lane >> 3;
        s = fminf(10.0f, fmaxf(-10.0f, s));
        const float ex = expf(s);
        expAtt[(size_t)e * HEAD + h] = ex;
        atomicAdd(&attNorm[(size_t)r * HEAD + h], ex);
    }
}

// ---------------------------------------------------------------------------
// Normalize + weighted scatter: one wave per edge. Reads raw expAtt from the
// d_out att region, normalizes in place, scatters att*V into out[row] with
// fp32 atomics (L2-side). Lane l owns dims [4l,4l+4), head = l/8.
// ---------------------------------------------------------------------------
__global__ __launch_bounds__(256) void gf_edge_scatter(
    const float* __restrict__ V,
    const int* __restrict__ rows, const int* __restrict__ cols,
    float* __restrict__ att,         // [E,4] raw exp in, normalized out
    const float* __restrict__ attNorm,
    float* __restrict__ out,         // [N,128]
    int n_edges)
{
    const int e    = blockIdx.x * (blockDim.x >> 5) + (threadIdx.x >> 5);
    const int lane = threadIdx.x & 31;
    if (e >= n_edges) return;

    const int r = rows[e];
    const int c = cols[e];
    const int h = lane >> 3;

    const float ex  = att[(size_t)e * HEAD + h];            // all lanes read first
    const float nrm = attNorm[(size_t)r * HEAD + h];
    const float a   = ex / (nrm + 1e-8f);

    if ((lane & 7) == 0) att[(size_t)e * HEAD + h] = a;     // in-place normalize

    const float4 vv = ((const float4*)(V + (size_t)c * DIM_E))[lane];
    float* orow = out + (size_t)r * DIM_E + lane * 4;
    atomicAdd(orow + 0, a * vv.x);
    atomicAdd(orow + 1, a * vv.y);
    atomicAdd(orow + 2, a * vv.z);
    atomicAdd(orow + 3, a * vv.w);
}

// ---------------------------------------------------------------------------
extern "C" void kernel_launch(void* const* d_in, const int* in_sizes, int n_in,
                              void* d_out, int out_size, void* d_ws, size_t ws_size,
                              hipStream_t stream)
{
    const float* embeds = (const float*)d_in[0];
    const float* qW     = (const float*)d_in[1];
    const float* kW     = (const float*)d_in[2];
    const float* vW     = (const float*)d_in[3];
    const int*   rows   = (const int*)d_in[4];
    const int*   cols   = (const int*)d_in[5];

    const int n_nodes = in_sizes[0] / DIM_E;   // embeds flat count / 128
    const int n_edges = in_sizes[4];

    // d_out = [ out: n_nodes*128 | att: n_edges*4 ]
    float* out = (float*)d_out;
    float* att = out + (size_t)n_nodes * DIM_E;

    // workspace: Q | K | V | attNorm
    float* Q       = (float*)d_ws;
    float* Kb      = Q  + (size_t)n_nodes * DIM_E;
    float* Vb      = Kb + (size_t)n_nodes * DIM_E;
    float* attNorm = Vb + (size_t)n_nodes * DIM_E;

    // 1) zero accumulation buffers
    gf_zero_f32<<<2048, 256, 0, stream>>>(out, (size_t)n_nodes * DIM_E);
    gf_zero_f32<<<256, 256, 0, stream>>>(attNorm, (size_t)n_nodes * HEAD);

    // 2) Q/K/V = embeds @ {qW,kW,vW}  (fp32 WMMA, one wave per 16x128 strip)
    dim3 ggrid((n_nodes + 15) / 16, 3);
    gf_qkv_gemm<<<ggrid, 32, 0, stream>>>(embeds, qW, kW, vW, Q, Kb, Vb, n_nodes);

    // 3) per-edge logits + softmax denominator (one wave per edge)
    const int wavesPerBlock = 8;
    const int eblocks = (n_edges + wavesPerBlock - 1) / wavesPerBlock;
    gf_edge_logits<<<eblocks, 32 * wavesPerBlock, 0, stream>>>(
        Q, Kb, rows, cols, att, attNorm, n_edges);

    // 4) normalize + weighted scatter-add of V
    gf_edge_scatter<<<eblocks, 32 * wavesPerBlock, 0, stream>>>(
        Vb, rows, cols, att, attNorm, out, n_edges);
}